// AdditiveAttentin_22256520528799
// MI455X (gfx1250) — compile-verified
//
#include <hip/hip_runtime.h>
#include <hip/hip_bf16.h>

typedef __attribute__((ext_vector_type(2))) float v2f;
typedef __attribute__((ext_vector_type(4))) float f4;
typedef __attribute__((ext_vector_type(8))) float v8f;

#define DD    128
#define BB    32
#define NN    6
#define ROWS  (BB * NN)                 // 192
#define SLICE (256 * 32 * 32)           // floats per (b,n) slice = 262144
#define OUT_ELEMS ((long long)BB * SLICE) // 8388608

// ---------------------------------------------------------------------------
// Kernel 1: fold W_out into W_feat/W_ctx, compute logits with
// V_WMMA_F32_16X16X4_F32, sparsemax over N=6 per batch.
// One block, 128 threads (4 waves). Tiny vs the streaming kernel.
// ---------------------------------------------------------------------------
__global__ __launch_bounds__(128) void attn_scores_kernel(
    const float* __restrict__ q,  const float* __restrict__ k,
    const float* __restrict__ Wf, const float* __restrict__ bf,
    const float* __restrict__ Wc, const float* __restrict__ bc,
    const float* __restrict__ Wo, const float* __restrict__ bo,
    float* __restrict__ attn_ws,  float* __restrict__ attn_out)
{
    __shared__ float w1[DD];
    __shared__ float w2[DD];
    __shared__ float sc[ROWS];

    const int tid = threadIdx.x;

    // ---- Phase A: w1 = W_out @ W_feat ; w2 = W_out @ W_ctx ; c0 -----------
    // One d-column per thread (128 threads == D). W_out[e]/bias loads are
    // loop-uniform -> scalar loads.
    float a1 = 0.f, a2 = 0.f, c0 = 0.f;
    for (int e = 0; e < DD; ++e) {
        float wo = Wo[e];                       // uniform
        c0 += wo * (bf[e] + bc[e]);             // uniform, redundant per lane
        a1 += wo * Wf[e * DD + tid];
        a2 += wo * Wc[e * DD + tid];
    }
    c0 += bo[0];
    w1[tid] = a1;
    w2[tid] = a2;
    __syncthreads();

    // ---- Phase B: scores[i] = w1.k_i + w2.q_i + c0 via WMMA ---------------
    // 12 M-tiles of 16 rows; each of the 4 waves handles 3 tiles.
    // A(16x4) f32 layout: lanes 0-15 hold K={0,1} in vgpr {0,1},
    //                     lanes 16-31 hold K={2,3}. B(4x16) mirrors this with
    // all 16 columns set to the same weight value, so every column of D is
    // the desired dot product.
    {
        const int l    = tid & 31;              // lane in wave
        const int wv   = tid >> 5;              // wave id 0..3
        const int m    = l & 15;                // row within tile
        const int koff = (l >> 4) * 2;          // +2 for upper half-wave

        for (int t = wv * 3; t < wv * 3 + 3; ++t) {
            const float* krow = k + (size_t)(t * 16 + m) * DD;
            const float* qrow = q + (size_t)(t * 16 + m) * DD;
            v8f c = {0.f, 0.f, 0.f, 0.f, 0.f, 0.f, 0.f, 0.f};
            for (int kc = 0; kc < DD; kc += 4) {
                v2f a, b;
                a.x = krow[kc + koff];     a.y = krow[kc + koff + 1];
                b.x = w1[kc + koff];       b.y = w1[kc + koff + 1];
                c = __builtin_amdgcn_wmma_f32_16x16x4_f32(
                        false, a, false, b, (short)0, c, false, false);
                a.x = qrow[kc + koff];     a.y = qrow[kc + koff + 1];
                b.x = w2[kc + koff];       b.y = w2[kc + koff + 1];
                c = __builtin_amdgcn_wmma_f32_16x16x4_f32(
                        false, a, false, b, (short)0, c, false, false);
            }
            // D layout: vgpr r, lanes 0-15 -> M=r ; lanes 16-31 -> M=r+8.
            // All N columns identical; lanes 0 and 16 publish their 8 rows.
            if ((l & 15) == 0) {
                const int mb = (l >> 4) * 8;
                #pragma unroll
                for (int r = 0; r < 8; ++r)
                    sc[t * 16 + mb + r] = c[r] + c0;
            }
        }
    }
    __syncthreads();

    // ---- Phase C: sparsemax over N=6, one batch per lane (wave 0) ---------
    if (tid < BB) {
        const int b = tid;
        float z[NN], zs[NN];
        #pragma unroll
        for (int n = 0; n < NN; ++n) { z[n] = sc[b * NN + n]; zs[n] = z[n]; }
        // descending sort (unrolled bubble network, stays in registers)
        #pragma unroll
        for (int i = 0; i < NN - 1; ++i) {
            #pragma unroll
            for (int j = 0; j < NN - 1 - i; ++j) {
                float lo = fminf(zs[j], zs[j + 1]);
                float hi = fmaxf(zs[j], zs[j + 1]);
                zs[j] = hi; zs[j + 1] = lo;
            }
        }
        // support size + cumsum at support boundary (reference semantics)
        float cs = 0.f, cs_k = 0.f;
        int kk = 0;
        #pragma unroll
        for (int j = 0; j < NN; ++j) {
            cs += zs[j];
            if (1.f + (float)(j + 1) * zs[j] > cs) { kk = j + 1; cs_k = cs; }
        }
        const float tau = (cs_k - 1.f) / (float)kk;
        #pragma unroll
        for (int n = 0; n < NN; ++n) {
            float p = fmaxf(z[n] - tau, 0.f);
            attn_ws[b * NN + n]  = p;   // consumed by streaming kernel
            attn_out[b * NN + n] = p;   // 2nd output: (B,1,N) flat == (B,N)
        }
    }
}

// ---------------------------------------------------------------------------
// Kernel 2: out[b,x] = sum_n attn[b,n] * v[b,n,x]   (x over C*H*W, float4)
// Pure HBM streaming: ~235 MB -> ~10us floor at 23.3 TB/s. Weights are
// block-uniform (scalar loads), so zero-weight slices are skipped by a
// wave-uniform branch — sparsemax sparsity directly cuts read traffic.
// ---------------------------------------------------------------------------
__global__ __launch_bounds__(256) void weighted_sum_kernel(
    const float* __restrict__ v, const float* __restrict__ attn_ws,
    float* __restrict__ out)
{
    const int  b  = blockIdx.y;
    const long long x4 = (long long)blockIdx.x * blockDim.x + threadIdx.x;
    const f4* vb = (const f4*)v + (long long)b * NN * (SLICE / 4);

    f4 acc = {0.f, 0.f, 0.f, 0.f};
    #pragma unroll
    for (int n = 0; n < NN; ++n) {
        const float wn = attn_ws[b * NN + n];        // uniform -> s_load
        if (wn != 0.f) {                             // uniform branch
            f4 vv = __builtin_nontemporal_load(&vb[(long long)n * (SLICE / 4) + x4]);
            acc.x += wn * vv.x; acc.y += wn * vv.y;
            acc.z += wn * vv.z; acc.w += wn * vv.w;
        }
    }
    __builtin_nontemporal_store(acc, (f4*)out + (long long)b * (SLICE / 4) + x4);
}

// ---------------------------------------------------------------------------
extern "C" void kernel_launch(void* const* d_in, const int* in_sizes, int n_in,
                              void* d_out, int out_size, void* d_ws, size_t ws_size,
                              hipStream_t stream)
{
    const float* q  = (const float*)d_in[0];
    const float* k  = (const float*)d_in[1];
    const float* v  = (const float*)d_in[2];
    const float* Wf = (const float*)d_in[3];
    const float* bf = (const float*)d_in[4];
    const float* Wc = (const float*)d_in[5];
    const float* bc = (const float*)d_in[6];
    const float* Wo = (const float*)d_in[7];
    const float* bo = (const float*)d_in[8];

    float* out      = (float*)d_out;
    float* attn_out = out + OUT_ELEMS;   // second tuple output, 192 floats
    float* attn_ws  = (float*)d_ws;      // 192-float scratch for kernel 2

    attn_scores_kernel<<<1, 128, 0, stream>>>(q, k, Wf, bf, Wc, bc, Wo, bo,
                                              attn_ws, attn_out);

    dim3 grid(SLICE / 4 / 256, BB);      // (256, 32)
    weighted_sum_kernel<<<grid, 256, 0, stream>>>(v, attn_ws, out);
}